// ConditionalPositionalEncoding1D_86071144611849
// MI455X (gfx1250) — compile-verified
//
#include <hip/hip_runtime.h>
#include <stdint.h>

// Problem constants (match reference setup_inputs)
#define B_    32
#define C_    256
#define T_    2048
// Tiling
#define ROWS  64          // rows (b,c) per block, one per thread
#define BLOCK 64          // threads per block (2 wave32)
#define TC    64          // time-chunk length (floats)
#define TCP   66          // padded LDS row stride: 8B-aligned, conflict-free wave32
#define NCH   (T_ / TC)   // 32 chunks
#define UNITS ((ROWS * TC / 2) / BLOCK)  // 32 b64 transfer units per thread per chunk

// CDNA5 split async counter wait (ASYNCcnt tracks global_load_async_to_lds)
#define ASYNC_WAIT_32() asm volatile("s_wait_asynccnt 0x20" ::: "memory")
#define ASYNC_WAIT_0()  asm volatile("s_wait_asynccnt 0x0"  ::: "memory")

// Async DMA: global -> LDS, 8 bytes per lane. VDST = LDS byte offset (VGPR),
// VADDR = 64-bit global address (VGPR pair), saddr = off (GV mode).
// The generic pointer's low 32 bits are the LDS byte offset (aperture layout),
// and we also pass the raw pointer so the shared array provably escapes.
__device__ __forceinline__ void issue_async_b64(const float* g, float* l) {
  unsigned off = (unsigned)(uintptr_t)l;
  asm volatile("global_load_async_to_lds_b64 %0, %1, off"
               :: "v"(off), "v"(g), "v"(l)
               : "memory");
}

// Issue one chunk of async loads: ROWS x TC floats, b64 per lane.
// Wave-coalesced: lanes 0..31 cover 256 contiguous bytes of one row.
__device__ __forceinline__ void issue_chunk(const float* __restrict__ x,
                                            int rowBase, int t0,
                                            float* lbuf, int tid) {
#pragma unroll
  for (int i = 0; i < UNITS; ++i) {
    const int u = tid + i * BLOCK;
    const int r = u >> 5;              // TC/2 == 32 b64 units per row
    const int w = (u & 31) << 1;       // float offset within chunk
    const float* g = x + (size_t)(rowBase + r) * T_ + t0 + w;
    issue_async_b64(g, &lbuf[r * TCP + w]);
  }
}

__global__ __launch_bounds__(BLOCK)
void cpe_lif_kernel(const float* __restrict__ x,
                    const float* __restrict__ conv_w,
                    const float* __restrict__ conv_b,
                    const float* __restrict__ gamma,
                    const float* __restrict__ beta,
                    const float* __restrict__ run_mean,
                    const float* __restrict__ run_var,
                    float* __restrict__ out) {
  __shared__ float lin[3 * ROWS * TCP];   // triple-buffered input tiles
  __shared__ float lout[ROWS * TCP];      // output staging tile

  const int tid     = threadIdx.x;
  const int rowBase = blockIdx.x * ROWS;
  const int row     = rowBase + tid;      // (b*C + c)
  const int c       = row & (C_ - 1);

  // Fold conv bias + BN into 3 weights and 1 bias per channel.
  const float inv  = gamma[c] * rsqrtf(run_var[c] + 1e-5f);
  const float W0   = conv_w[3 * c + 0] * inv;
  const float W1   = conv_w[3 * c + 1] * inv;
  const float W2   = conv_w[3 * c + 2] * inv;
  const float Bias = conv_b[c] * inv + beta[c] - run_mean[c] * inv;

  // Prime the pipeline: chunks 0 and 1 in flight.
  issue_chunk(x, rowBase, 0,  &lin[0],          tid);
  issue_chunk(x, rowBase, TC, &lin[ROWS * TCP], tid);

  float v  = 0.0f;   // LIF membrane
  float xp = 0.0f;   // x[t-1] (left halo carried in register; t=-1 pad = 0)
  const int rb = tid * TCP;

  for (int k = 0; k < NCH; ++k) {
    const int t0 = k * TC;
    const float* linK  = &lin[(k % 3) * (ROWS * TCP)];
    const float* linK1 = &lin[((k + 1) % 3) * (ROWS * TCP)];

    if (k + 2 < NCH) {
      // Overwrites buffer of chunk k-1 (fully consumed before last barrier).
      issue_chunk(x, rowBase, t0 + 2 * TC, &lin[((k + 2) % 3) * (ROWS * TCP)], tid);
      // Loads retire in order: <=32 outstanding => chunks k and k+1 resident.
      ASYNC_WAIT_32();
    } else {
      ASYNC_WAIT_0();
    }
    __syncthreads();

    // Sequential LIF scan over this chunk; conflict-free LDS reads (bank 2r).
    float xc = linK[rb];
#pragma unroll 7
    for (int t = 0; t < TC - 1; ++t) {
      const float xn = linK[rb + t + 1];
      const float y  = fmaf(W0, xp, fmaf(W1, xc, fmaf(W2, xn, Bias)));
      v += (y - v) * 0.5f;
      const float s = (v >= 1.0f) ? 1.0f : 0.0f;
      lout[rb + t] = s + xc;                 // spike + residual
      v = (v >= 1.0f) ? 0.0f : v;            // detached hard reset
      xp = xc; xc = xn;
    }
    { // last step: right halo = first element of next chunk (or 0 at t=T-1)
      const float xn = (k + 1 < NCH) ? linK1[rb] : 0.0f;
      const float y  = fmaf(W0, xp, fmaf(W1, xc, fmaf(W2, xn, Bias)));
      v += (y - v) * 0.5f;
      const float s = (v >= 1.0f) ? 1.0f : 0.0f;
      lout[rb + TC - 1] = s + xc;
      v = (v >= 1.0f) ? 0.0f : v;
      xp = xc;
    }
    __syncthreads();

    // Coalesced writeback: lanes 0..31 cover 256 contiguous bytes of one row.
#pragma unroll
    for (int i = 0; i < UNITS; ++i) {
      const int u = tid + i * BLOCK;
      const int r = u >> 5;
      const int w = (u & 31) << 1;
      const float2 d = *(const float2*)&lout[r * TCP + w];
      *(float2*)&out[(size_t)(rowBase + r) * T_ + t0 + w] = d;
    }
    // No extra barrier: next iteration's lout writes happen only after its
    // first __syncthreads, and these LDS reads are consumed before the stores.
  }
}

extern "C" void kernel_launch(void* const* d_in, const int* in_sizes, int n_in,
                              void* d_out, int out_size, void* d_ws, size_t ws_size,
                              hipStream_t stream) {
  const float* x        = (const float*)d_in[0];
  const float* conv_w   = (const float*)d_in[1];
  const float* conv_b   = (const float*)d_in[2];
  const float* gamma    = (const float*)d_in[3];
  const float* beta     = (const float*)d_in[4];
  const float* run_mean = (const float*)d_in[5];
  const float* run_var  = (const float*)d_in[6];
  float* out            = (float*)d_out;

  const int grid = (B_ * C_) / ROWS;   // 128 blocks
  cpe_lif_kernel<<<grid, BLOCK, 0, stream>>>(x, conv_w, conv_b, gamma, beta,
                                             run_mean, run_var, out);
}